// AdvancedLossFunction_77326591197732
// MI455X (gfx1250) — compile-verified
//
#include <hip/hip_runtime.h>
#include <hip/hip_bf16.h>

typedef float v2f __attribute__((ext_vector_type(2)));
typedef float v8f __attribute__((ext_vector_type(8)));
typedef unsigned long long u64;

#define N_PTS   16384
#define F_DIM   64
#define LEPS    1e-7f
#define CHUNK   128    // db points staged in LDS per block iteration
#define NSEG    4      // database split for occupancy
#define SEG_LEN (N_PTS / NSEG)
#define NCHUNK  (SEG_LEN / CHUNK)

// ---------------------------------------------------------------------------
// Kernel A: pack points to float4 (w=0) and compute squared norms
// ---------------------------------------------------------------------------
__global__ void pack_points_kernel(const float* __restrict__ points,
                                   float* __restrict__ packed,
                                   float* __restrict__ sqv) {
  int i = blockIdx.x * blockDim.x + threadIdx.x;
  if (i < N_PTS) {
    float x = points[i * 3 + 0];
    float y = points[i * 3 + 1];
    float z = points[i * 3 + 2];
    packed[i * 4 + 0] = x;
    packed[i * 4 + 1] = y;
    packed[i * 4 + 2] = z;
    packed[i * 4 + 3] = 0.0f;
    sqv[i] = x * x + y * y + z * z;
  }
}

// ---------------------------------------------------------------------------
// Kernel B: BCE + MSE block partials (64 blocks x 256 threads = 16384)
// ---------------------------------------------------------------------------
__global__ void bce_mse_kernel(const float* __restrict__ pred,
                               const float* __restrict__ targ,
                               float* __restrict__ part_bce,
                               float* __restrict__ part_mse) {
  __shared__ float s1[256];
  __shared__ float s2[256];
  int i = blockIdx.x * 256 + threadIdx.x;
  float p = pred[i];
  float t = targ[i];
  float pc = fminf(fmaxf(p, LEPS), 1.0f - LEPS);
  float bce = t * logf(pc) + (1.0f - t) * log1pf(-pc);
  float d = p - t;
  s1[threadIdx.x] = bce;
  s2[threadIdx.x] = d * d;
  __syncthreads();
  for (int off = 128; off > 0; off >>= 1) {
    if (threadIdx.x < off) {
      s1[threadIdx.x] += s1[threadIdx.x + off];
      s2[threadIdx.x] += s2[threadIdx.x + off];
    }
    __syncthreads();
  }
  if (threadIdx.x == 0) {
    part_bce[blockIdx.x] = s1[0];
    part_mse[blockIdx.x] = s2[0];
  }
}

// ---------------------------------------------------------------------------
// Kernel C: mean|features| partials (4096 blocks x 256 threads = N*F)
// ---------------------------------------------------------------------------
__global__ void feat_abs_kernel(const float* __restrict__ feat,
                                float* __restrict__ part_feat) {
  __shared__ float s1[256];
  int i = blockIdx.x * 256 + threadIdx.x;
  s1[threadIdx.x] = fabsf(feat[i]);
  __syncthreads();
  for (int off = 128; off > 0; off >>= 1) {
    if (threadIdx.x < off) s1[threadIdx.x] += s1[threadIdx.x + off];
    __syncthreads();
  }
  if (threadIdx.x == 0) part_feat[blockIdx.x] = s1[0];
}

// ---------------------------------------------------------------------------
// Top-4 via monotonic u64 keys: key = (float_bits(d>=0) << 32) | index.
// Unsigned order on the key == lexicographic (d, idx) — identical to JAX's
// stable top_k. Insertion into a sorted 4-list is a pure min/max network
// (native v_min_u64/v_max_u64 on gfx1250 — fully branch-free).
// ---------------------------------------------------------------------------
__device__ __forceinline__ void insert4_key(u64 k[4], u64 x) {
  k[3] = __builtin_elementwise_min(k[3], __builtin_elementwise_max(k[2], x));
  k[2] = __builtin_elementwise_min(k[2], __builtin_elementwise_max(k[1], x));
  k[1] = __builtin_elementwise_min(k[1], __builtin_elementwise_max(k[0], x));
  k[0] = __builtin_elementwise_min(k[0], x);
}

// Candidate consumption for one 16-row WMMA tile; SELF variant only runs for
// the single (wave-uniform) tile that contains the query's own row.
template <bool SELF>
__device__ __forceinline__ void consume_tile(const v8f& c, int mgBase, int half,
                                             int q, float sqq,
                                             const float* __restrict__ srow,
                                             u64 best[4]) {
#pragma unroll
  for (int r = 0; r < 8; ++r) {
    int mloc = half * 8 + r;
    int mg   = mgBase + mloc;
    float d2 = fmaf(-2.0f, c[r], srow[mloc] + sqq);
    float d  = fmaxf(d2, 0.0f);            // d >= 0: bits are monotone
    u64 key = ((u64)__float_as_uint(d) << 32) | (unsigned)mg;
    if (SELF) key = (mg == q) ? 0ull : key;  // self forced to global rank 0
    insert4_key(best, key);
  }
}

// ---------------------------------------------------------------------------
// Kernel D: per-segment streaming 3-NN via V_WMMA_F32_16X16X4_F32.
// Grid (256, NSEG) x 128 threads (4 waves). Each wave owns 16 query points
// (one query column per lane in the WMMA C layout) and sweeps one 4096-point
// db segment staged through double-buffered async load-to-LDS (ASYNCcnt):
// chunk i+1 is in flight while chunk i is consumed. Each lane keeps a sorted
// top-4 key list in registers; lane n / n+16 merged via shfl; the per-segment
// sorted top-4 keys go to workspace.
// ---------------------------------------------------------------------------
__global__ void knn_segment_kernel(const float* __restrict__ packed,
                                   const float* __restrict__ sqv,
                                   u64* __restrict__ partk) {
  __shared__ __align__(16) float ldsP[2][CHUNK * 4];
  __shared__ float ldsS[2][CHUNK];

  int tid  = threadIdx.x;
  int lane = tid & 31;
  int wave = tid >> 5;
  int half = lane >> 4;   // 0: rows M=0..7 ; 1: rows M=8..15
  int lp   = lane & 15;   // query column within the wave's 16 queries

  int q0 = blockIdx.x * 64 + wave * 16;
  int q  = q0 + lp;
  int segBase = blockIdx.y * SEG_LEN;

  // B matrix (4x16 queries): lanes 0-15 hold K=0,1 ; lanes 16-31 hold K=2,3
  v2f B;
  B.x = packed[q * 4 + half * 2 + 0];
  B.y = packed[q * 4 + half * 2 + 1];
  float sqq = sqv[q];

  // LDS byte offsets for async load-to-LDS (low 32 bits of flat LDS address)
  unsigned pAddr[2], sAddr[2];
#pragma unroll
  for (int b = 0; b < 2; ++b) {
    pAddr[b] = (unsigned)(uintptr_t)(void*)&ldsP[b][0] + (unsigned)tid * 16u;
    sAddr[b] = (unsigned)(uintptr_t)(void*)&ldsS[b][0] + (unsigned)tid * 4u;
  }

  u64 best[4] = {~0ull, ~0ull, ~0ull, ~0ull};

  // issue chunk 0 into buffer 0
  {
    const float4* gP = (const float4*)packed + (segBase + tid);
    const float*  gS = sqv + (segBase + tid);
    asm volatile("global_load_async_to_lds_b128 %0, %1, off"
                 :: "v"(pAddr[0]), "v"(gP) : "memory");
    asm volatile("global_load_async_to_lds_b32 %0, %1, off"
                 :: "v"(sAddr[0]), "v"(gS) : "memory");
  }

  for (int ci = 0; ci < NCHUNK; ++ci) {
    int buf  = ci & 1;
    int base = segBase + ci * CHUNK;
    if (ci + 1 < NCHUNK) {
      // issue next chunk into the other buffer (2 async ops per wave)
      int nbase = base + CHUNK;
      const float4* gP = (const float4*)packed + (nbase + tid);
      const float*  gS = sqv + (nbase + tid);
      asm volatile("global_load_async_to_lds_b128 %0, %1, off"
                   :: "v"(pAddr[buf ^ 1]), "v"(gP) : "memory");
      asm volatile("global_load_async_to_lds_b32 %0, %1, off"
                   :: "v"(sAddr[buf ^ 1]), "v"(gS) : "memory");
      // ASYNCcnt completes in order: <=2 left means chunk ci has landed
      asm volatile("s_wait_asynccnt 0x2" ::: "memory");
      if (ci + 2 < NCHUNK)
        __builtin_prefetch(&packed[(base + 2 * CHUNK + tid) * 4], 0, 3);
    } else {
      asm volatile("s_wait_asynccnt 0x0" ::: "memory");
    }
    __syncthreads();   // publish chunk ci's LDS data to all waves

#pragma unroll
    for (int t16 = 0; t16 < CHUNK / 16; ++t16) {
      int dbOff = t16 * 16;
      // A matrix (16x4 db points): identical lane striping to B
      v2f A;
      A.x = ldsP[buf][(dbOff + lp) * 4 + half * 2 + 0];
      A.y = ldsP[buf][(dbOff + lp) * 4 + half * 2 + 1];
      v8f c = {};
      c = __builtin_amdgcn_wmma_f32_16x16x4_f32(
          /*neg_a=*/false, A, /*neg_b=*/false, B,
          /*c_mod=*/(short)0, c, /*reuse_a=*/false, /*reuse_b=*/false);
      const float* srow = &ldsS[buf][dbOff];
      // wave-uniform: this tile contains the wave's own 16 query rows
      if (base + dbOff == q0)
        consume_tile<true>(c, base + dbOff, half, q, sqq, srow, best);
      else
        consume_tile<false>(c, base + dbOff, half, q, sqq, srow, best);
    }
    __syncthreads();   // all waves done reading buf before it is re-filled
  }

  // merge lane n (rows 0-7 of each tile) with lane n+16 (rows 8-15)
  u64 other[4];
#pragma unroll
  for (int k = 0; k < 4; ++k) other[k] = __shfl(best[k], lp + 16, 32);
  if (half == 0) {
#pragma unroll
    for (int k = 0; k < 4; ++k) insert4_key(best, other[k]);
    int outBase = (blockIdx.y * N_PTS + q) * 4;
#pragma unroll
    for (int k = 0; k < 4; ++k) partk[outBase + k] = best[k];
  }
}

// ---------------------------------------------------------------------------
// Kernel E: merge the NSEG sorted per-query key lists, compute smoothness
// partials (64 blocks x 256 threads = 16384 queries)
// ---------------------------------------------------------------------------
__global__ void merge_smooth_kernel(const u64* __restrict__ partk,
                                    const float* __restrict__ pred,
                                    float* __restrict__ part_sm) {
  __shared__ float s1[256];
  int q = blockIdx.x * 256 + threadIdx.x;

  u64 best[4];
#pragma unroll
  for (int k = 0; k < 4; ++k) best[k] = partk[q * 4 + k];
#pragma unroll
  for (int seg = 1; seg < NSEG; ++seg) {
    int base = (seg * N_PTS + q) * 4;
#pragma unroll
    for (int k = 0; k < 4; ++k) insert4_key(best, partk[base + k]);
  }
  // rank 0 is self (key forced to 0); neighbors are ranks 1..3
  int n1 = (int)(unsigned)(best[1] & 0xFFFFFFFFull);
  int n2 = (int)(unsigned)(best[2] & 0xFFFFFFFFull);
  int n3 = (int)(unsigned)(best[3] & 0xFFFFFFFFull);
  float pq = pred[q];
  float s = fabsf(pq - pred[n1]) +
            fabsf(pq - pred[n2]) +
            fabsf(pq - pred[n3]);
  s1[threadIdx.x] = s;
  __syncthreads();
  for (int off = 128; off > 0; off >>= 1) {
    if (threadIdx.x < off) s1[threadIdx.x] += s1[threadIdx.x + off];
    __syncthreads();
  }
  if (threadIdx.x == 0) part_sm[blockIdx.x] = s1[0];
}

// ---------------------------------------------------------------------------
// Kernel F: deterministic serial finalize
// ---------------------------------------------------------------------------
__global__ void finalize_kernel(const float* __restrict__ part_bce,
                                const float* __restrict__ part_mse,
                                const float* __restrict__ part_feat,
                                const float* __restrict__ part_sm,
                                float* __restrict__ out) {
  if (threadIdx.x == 0 && blockIdx.x == 0) {
    float bce = 0.0f, mse = 0.0f, fsum = 0.0f, sm = 0.0f;
    for (int i = 0; i < 64; ++i) bce += part_bce[i];
    for (int i = 0; i < 64; ++i) mse += part_mse[i];
    for (int i = 0; i < 4096; ++i) fsum += part_feat[i];
    for (int i = 0; i < 64; ++i) sm += part_sm[i];
    float occupancy   = -bce / (float)N_PTS;
    float smoothness  = sm / (float)(N_PTS * 3);
    float sparsity    = fsum / (float)(N_PTS * F_DIM);
    float consistency = mse / (float)N_PTS;
    out[0] = 1.0f * occupancy + 0.1f * smoothness +
             0.01f * sparsity + 0.1f * consistency;
  }
}

// ---------------------------------------------------------------------------
extern "C" void kernel_launch(void* const* d_in, const int* in_sizes, int n_in,
                              void* d_out, int out_size, void* d_ws, size_t ws_size,
                              hipStream_t stream) {
  const float* pred = (const float*)d_in[0];   // predictions (16384)
  const float* targ = (const float*)d_in[1];   // targets     (16384)
  const float* feat = (const float*)d_in[2];   // features    (16384*64)
  const float* pts  = (const float*)d_in[3];   // points      (16384*3)

  float* ws        = (float*)d_ws;
  float* packed    = ws;                        // 16384*4 floats
  float* sqv       = packed + N_PTS * 4;        // 16384
  float* part_feat = sqv + N_PTS;               // 4096
  float* part_bce  = part_feat + 4096;          // 64
  float* part_mse  = part_bce + 64;             // 64
  float* part_sm   = part_mse + 64;             // 64
  // 8-byte aligned region for keys (offset is a multiple of 8 floats)
  u64*   partk     = (u64*)(part_sm + 64);      // NSEG*16384*4 u64

  pack_points_kernel<<<64, 256, 0, stream>>>(pts, packed, sqv);
  bce_mse_kernel<<<64, 256, 0, stream>>>(pred, targ, part_bce, part_mse);
  feat_abs_kernel<<<4096, 256, 0, stream>>>(feat, part_feat);
  knn_segment_kernel<<<dim3(256, NSEG), 128, 0, stream>>>(packed, sqv, partk);
  merge_smooth_kernel<<<64, 256, 0, stream>>>(partk, pred, part_sm);
  finalize_kernel<<<1, 1, 0, stream>>>(part_bce, part_mse, part_feat, part_sm,
                                       (float*)d_out);
}